// TopologicalFeatureExtractor_34402688041035
// MI455X (gfx1250) — compile-verified
//
#include <hip/hip_runtime.h>

#define NC   10
#define HH   128
#define WW   128
#define NPIX (HH * WW)
#define NB   128
#define RAWF 60
#define CCL_THREADS 1024

typedef __attribute__((ext_vector_type(2))) float v2f;
typedef __attribute__((ext_vector_type(8))) float v8f;

// ---------------- Union-find on LDS labels (one 128x128 image per workgroup) -------
// Parents are monotone decreasing (attach-by-min), so lock-free atomicMin linking is
// safe; volatile loads stop the compiler from register-caching spin reads. All ops
// are ds_* (LDS) — no L2 round trips.

__device__ __forceinline__ int find_root(int* L, int x) {
  for (;;) {
    int p = *(volatile int*)&L[x];
    if (p == x) return x;
    int gp = *(volatile int*)&L[p];
    if (gp == p) return p;
    *(volatile int*)&L[x] = gp;   // path halving (benign race: gp is a valid ancestor)
    x = gp;
  }
}

__device__ __forceinline__ void unite(int* L, int a, int b) {
  int ra = find_root(L, a);
  int rb = find_root(L, b);
  while (ra != rb) {
    if (ra < rb) { int t = ra; ra = rb; rb = t; }   // ra = larger
    int old = atomicMin(&L[ra], rb);
    if (old == ra) return;                          // linked ra -> rb
    ra = find_root(L, old);
    rb = find_root(L, rb);
  }
}

// ---------------- Kernel 1: CCL + per-color topological features ----------------
// One block per batch image; 1024 threads (32 wave32s). Labels + colors in LDS
// (64KB + 16KB + 384B stats = ~82KB of the WGP's 320KB).

__global__ __launch_bounds__(CCL_THREADS) void ccl_feats_kernel(
    const int* __restrict__ grid, float* __restrict__ feats) {
  __shared__ int L[NPIX];                 // 64 KB labels
  __shared__ unsigned char col[NPIX];     // 16 KB colors
  __shared__ int sStat[6][16];            // 0:V 1:E 2:beta0 3:mismH 4:mismV 5:mismR

  const int b = blockIdx.x;
  const int tid = (int)threadIdx.x;
  const int* g = grid + (size_t)b * NPIX;

  if (tid < 96) ((int*)sStat)[tid] = 0;
  for (int i = tid; i < NPIX; i += CCL_THREADS) {
    L[i] = i;
    col[i] = (unsigned char)g[i];         // single coalesced pass over global grid
  }
  __syncthreads();

  // Merge right / down same-color neighbors (covers full 4-adjacency).
  for (int i = tid; i < NPIX; i += CCL_THREADS) {
    const int y = i >> 7, x = i & (WW - 1);
    const int c = col[i];
    if (x + 1 < WW && col[i + 1]  == c) unite(L, i, i + 1);
    if (y + 1 < HH && col[i + WW] == c) unite(L, i, i + WW);
  }
  __syncthreads();

  // Per-pixel stats.
  for (int i = tid; i < NPIX; i += CCL_THREADS) {
    const int y = i >> 7, x = i & (WW - 1);
    const int c = col[i];
    const int ci = c & 15;
    atomicAdd(&sStat[0][ci], 1);                                        // V
    if (x + 1 < WW && col[i + 1]  == c) atomicAdd(&sStat[1][ci], 1);    // E (horiz)
    if (y + 1 < HH && col[i + WW] == c) atomicAdd(&sStat[1][ci], 1);    // E (vert)
    if (find_root(L, i) == i) atomicAdd(&sStat[2][ci], 1);              // beta0: roots
    const int chh = col[y * WW + (WW - 1 - x)];
    if (chh != c) { atomicAdd(&sStat[3][ci], 1); atomicAdd(&sStat[3][chh & 15], 1); }
    const int cvv = col[(HH - 1 - y) * WW + x];
    if (cvv != c) { atomicAdd(&sStat[4][ci], 1); atomicAdd(&sStat[4][cvv & 15], 1); }
    const int crr = col[(HH - 1 - y) * WW + (WW - 1 - x)];
    if (crr != c) { atomicAdd(&sStat[5][ci], 1); atomicAdd(&sStat[5][crr & 15], 1); }
  }
  __syncthreads();

  if (tid < NC) {
    const float V  = (float)sStat[0][tid];
    const float E  = (float)sStat[1][tid];
    const float b0 = (float)sStat[2][tid];
    const float b1 = fmaxf(0.0f, b0 - (V - E));
    const float inv = 1.0f / (float)NPIX;
    const float hs = 1.0f - (float)sStat[3][tid] * inv;
    const float vs = 1.0f - (float)sStat[4][tid] * inv;
    const float rs = 1.0f - (float)sStat[5][tid] * inv;
    const float perim = 4.0f * V - 2.0f * E;
    float comp;
    if (V == 0.0f)            comp = 0.0f;
    else if (perim == 0.0f)   comp = 1.0f;
    else                      comp = fminf(1.0f, 12.566370614359172f * V / (perim * perim));
    float* f = feats + b * RAWF + tid * 6;
    f[0] = b0; f[1] = b1; f[2] = hs; f[3] = vs; f[4] = rs; f[5] = comp;
  }
}

// ---------------- Kernel 2: fused MLP with V_WMMA_F32_16X16X4_F32 ----------------
// One block, 8 waves. M=128 always; N in {128,64,32}; K in {60,128,64} (all %4==0).

__device__ __forceinline__ void gemm_wmma(const float* __restrict__ X, int K,
                                          const float* __restrict__ Wm,
                                          const float* __restrict__ bias, int N,
                                          float* __restrict__ Y, bool relu,
                                          int lane, int wave) {
  const int ntiles = N >> 4;
  const int total  = (NB / 16) * ntiles;   // M tiles (8) x N tiles
  const int mlane  = lane & 15;
  const int khalf  = (lane >> 4) << 1;     // lanes 16-31 hold K-half {2,3}
  for (int t = wave; t < total; t += 8) {
    const int mt = t / ntiles;
    const int nt = t - mt * ntiles;
    const int arow = mt * 16 + mlane;      // A: M = lane%16
    const int bcol = nt * 16 + mlane;      // B: N = lane%16
    v8f acc = {0.f, 0.f, 0.f, 0.f, 0.f, 0.f, 0.f, 0.f};
    for (int k0 = 0; k0 < K; k0 += 4) {
      const int ka = k0 + khalf;
      v2f af, bf;
      af.x = X[arow * K + ka];             // A VGPR0: K = khalf
      af.y = X[arow * K + ka + 1];         // A VGPR1: K = khalf+1
      bf.x = Wm[ka * N + bcol];            // B VGPR0
      bf.y = Wm[(ka + 1) * N + bcol];      // B VGPR1
      acc = __builtin_amdgcn_wmma_f32_16x16x4_f32(false, af, false, bf,
                                                  (short)0, acc, false, false);
    }
    // C/D layout: VGPR r -> row r (lanes 0-15) / row r+8 (lanes 16-31), col = lane%16
    const int nOut  = nt * 16 + mlane;
    const int mBase = mt * 16 + ((lane >> 4) << 3);
    const float bv = bias[nOut];
#pragma unroll
    for (int r = 0; r < 8; ++r) {
      float v = acc[r] + bv;
      if (relu) v = fmaxf(v, 0.0f);
      Y[(mBase + r) * N + nOut] = v;
    }
  }
}

__device__ __forceinline__ void layernorm_rows(float* __restrict__ Y, int N,
                                               const float* __restrict__ gamma,
                                               const float* __restrict__ beta,
                                               int lane, int wave) {
  const int per = N >> 5;                  // elems per lane (wave32)
  for (int row = wave; row < NB; row += 8) {
    float vals[4];
    float s = 0.0f;
    for (int j = 0; j < per; ++j) { vals[j] = Y[row * N + j * 32 + lane]; s += vals[j]; }
    for (int off = 16; off > 0; off >>= 1) s += __shfl_xor(s, off, 32);
    const float mu = s / (float)N;
    float v = 0.0f;
    for (int j = 0; j < per; ++j) { float d = vals[j] - mu; v += d * d; }
    for (int off = 16; off > 0; off >>= 1) v += __shfl_xor(v, off, 32);
    const float rinv = rsqrtf(v / (float)N + 1e-5f);
    for (int j = 0; j < per; ++j) {
      const int n = j * 32 + lane;
      Y[row * N + n] = (vals[j] - mu) * rinv * gamma[n] + beta[n];
    }
  }
}

__global__ __launch_bounds__(256) void mlp_wmma_kernel(
    const float* __restrict__ feats,
    const float* __restrict__ w1, const float* __restrict__ b1,
    const float* __restrict__ g1, const float* __restrict__ be1,
    const float* __restrict__ w2, const float* __restrict__ b2,
    const float* __restrict__ g2, const float* __restrict__ be2,
    const float* __restrict__ w3, const float* __restrict__ b3,
    float* __restrict__ h1, float* __restrict__ h2, float* __restrict__ out) {
  const int lane = (int)threadIdx.x & 31;
  const int wave = (int)threadIdx.x >> 5;
  gemm_wmma(feats, RAWF, w1, b1, 128, h1, true, lane, wave);
  __syncthreads();
  layernorm_rows(h1, 128, g1, be1, lane, wave);
  __syncthreads();
  gemm_wmma(h1, 128, w2, b2, 64, h2, true, lane, wave);
  __syncthreads();
  layernorm_rows(h2, 64, g2, be2, lane, wave);
  __syncthreads();
  gemm_wmma(h2, 64, w3, b3, 32, out, false, lane, wave);
}

// ---------------- launcher ----------------
extern "C" void kernel_launch(void* const* d_in, const int* in_sizes, int n_in,
                              void* d_out, int out_size, void* d_ws, size_t ws_size,
                              hipStream_t stream) {
  const int*   grid = (const int*)d_in[0];
  const float* w1   = (const float*)d_in[1];
  const float* b1   = (const float*)d_in[2];
  const float* g1   = (const float*)d_in[3];
  const float* be1  = (const float*)d_in[4];
  const float* w2   = (const float*)d_in[5];
  const float* b2   = (const float*)d_in[6];
  const float* g2   = (const float*)d_in[7];
  const float* be2  = (const float*)d_in[8];
  const float* w3   = (const float*)d_in[9];
  const float* b3   = (const float*)d_in[10];
  float* out = (float*)d_out;

  const int B = in_sizes[0] / NPIX;   // 128

  // Workspace: feats [B,60] | h1 [128,128] | h2 [128,64]  (~130 KB total)
  char* ws = (char*)d_ws;
  float* feats = (float*)ws;
  float* h1    = (float*)(ws + 32768);            // after feats (30720B, aligned up)
  float* h2    = (float*)(ws + 32768 + 65536);

  ccl_feats_kernel<<<B, CCL_THREADS, 0, stream>>>(grid, feats);
  mlp_wmma_kernel<<<1, 256, 0, stream>>>(feats, w1, b1, g1, be1, w2, b2, g2, be2,
                                         w3, b3, h1, h2, out);
}